// SSIMLoss_28948079575177
// MI455X (gfx1250) — compile-verified
//
#include <hip/hip_runtime.h>

typedef float v2f __attribute__((ext_vector_type(2)));
typedef float v4f __attribute__((ext_vector_type(4)));
typedef float v8f __attribute__((ext_vector_type(8)));

#define IMG_H 512
#define IMG_W 512
#define OUT_H 502
#define OUT_W 502
#define NIMG  32
#define TILE  16
#define ITILE 26      // TILE + 2*5 halo
#define IPITCH 28     // padded pitch (even, conflict-free mod 64 banks)
#define CHUNKS 7      // K padded 26 -> 28, 7 chunks of K=4
#define TPD   32      // tiles per dim: ceil(502/16)
#define NTILES (NIMG * TPD * TPD)   // 32768 partial sums

// 11-tap Gaussian, sigma = 1.5, normalized (matches reference _gauss_1d)
__device__ __constant__ float c_g[11] = {
    0.00102838f, 0.00759877f, 0.03600077f, 0.10936069f, 0.21300553f,
    0.26601166f,
    0.21300553f, 0.10936069f, 0.03600077f, 0.00759877f, 0.00102838f
};

#define WMMA_F32X4(A, B, C) \
    __builtin_amdgcn_wmma_f32_16x16x4_f32(false, (A), false, (B), (short)0, (C), false, false)

__device__ __forceinline__ void store_frag_rows(float* dst, v8f a) {
    // dst points at lh[q][j][mb]; 8 consecutive row entries, 8B aligned
    v2f t;
    t.x = a[0]; t.y = a[1]; *(v2f*)(dst + 0) = t;
    t.x = a[2]; t.y = a[3]; *(v2f*)(dst + 2) = t;
    t.x = a[4]; t.y = a[5]; *(v2f*)(dst + 4) = t;
    t.x = a[6]; t.y = a[7]; *(v2f*)(dst + 6) = t;
}

__global__ __launch_bounds__(32)
void ssim_tile_kernel(const float* __restrict__ inA,   // y_pred
                      const float* __restrict__ inB,   // y_true
                      float* __restrict__ partial) {
    __shared__ float lx[ITILE][IPITCH];
    __shared__ float ly[ITILE][IPITCH];
    __shared__ float lh[5][TILE][IPITCH];  // transposed: [quantity][col j][row k]

    const int lane = threadIdx.x;           // one wave per block
    const int bid  = blockIdx.x;
    const int tx   = bid & (TPD - 1);
    const int ty   = (bid >> 5) & (TPD - 1);
    const int img  = bid >> 10;
    const int c0   = tx * TILE;
    const int r0   = ty * TILE;

    const float* baseA = inA + (size_t)img * IMG_H * IMG_W;
    const float* baseB = inB + (size_t)img * IMG_H * IMG_W;

    // ---- stage 26x28 input tiles to LDS ----
    if (r0 + ITILE + 1 < IMG_H && c0 + IPITCH - 1 < IMG_W) {
        // interior tile: vectorized float4 path (global & LDS 16B aligned)
        for (int idx = lane; idx < ITILE * (IPITCH / 4); idx += 32) {
            int r = idx / (IPITCH / 4), c4 = (idx % (IPITCH / 4)) * 4;
            size_t goff = (size_t)(r0 + r) * IMG_W + (c0 + c4);
            *(v4f*)&lx[r][c4] = *(const v4f*)&baseA[goff];
            *(v4f*)&ly[r][c4] = *(const v4f*)&baseB[goff];
        }
    } else {
        // edge tile: clamped scalar path (values past the edge are masked later)
        for (int idx = lane; idx < ITILE * IPITCH; idx += 32) {
            int r = idx / IPITCH, c = idx % IPITCH;
            int gr = min(r0 + r, IMG_H - 1);
            int gc = min(c0 + c, IMG_W - 1);
            lx[r][c] = baseA[gr * IMG_W + gc];
            ly[r][c] = baseB[gr * IMG_W + gc];
        }
    }
    // zero rows 26,27 of every lh plane (zero-weight tail of K-chunk 6)
    for (int idx = lane; idx < 5 * TILE * 2; idx += 32) {
        int q = idx >> 5, rest = idx & 31;
        lh[q][rest >> 1][ITILE + (rest & 1)] = 0.0f;
    }
    __syncthreads();

    const int jrow  = lane & 15;        // N (or M) index carried by this lane
    const int khalf = (lane >> 4) * 2;  // K sub-offset for this lane half

    // ---- banded Gaussian fragment: value(k, i) = g[k-i] if 0<=k-i<=10 ----
    // Identical layout whether used as horizontal-B or vertical-A operand.
    v2f wfrag[CHUNKS];
#pragma unroll
    for (int kc = 0; kc < CHUNKS; ++kc) {
        int k0 = 4 * kc + khalf;
        int d0 = k0 - jrow;
        int d1 = d0 + 1;
        wfrag[kc].x = (d0 >= 0 && d0 <= 10) ? c_g[d0] : 0.0f;
        wfrag[kc].y = (d1 >= 0 && d1 <= 10) ? c_g[d1] : 0.0f;
    }

    // ---- horizontal pass: shared A-loads feed all 5 quantities ----
#pragma unroll
    for (int rb = 0; rb < 2; ++rb) {
        const int rowbase = rb ? 10 : 0;       // rows 0..15 then 10..25
        const int r = rowbase + jrow;          // A-matrix M = lane & 15
        v8f h0 = {}, h1 = {}, h2 = {}, h3 = {}, h4 = {};
#pragma unroll
        for (int kc = 0; kc < CHUNKS; ++kc) {
            int cb = 4 * kc + khalf;
            v2f xv = *(const v2f*)&lx[r][cb];
            v2f yv = *(const v2f*)&ly[r][cb];
            v2f axx, ayy, axy;
            axx.x = xv.x * xv.x; axx.y = xv.y * xv.y;
            ayy.x = yv.x * yv.x; ayy.y = yv.y * yv.y;
            axy.x = xv.x * yv.x; axy.y = xv.y * yv.y;
            h0 = WMMA_F32X4(xv,  wfrag[kc], h0);
            h1 = WMMA_F32X4(yv,  wfrag[kc], h1);
            h2 = WMMA_F32X4(axx, wfrag[kc], h2);
            h3 = WMMA_F32X4(ayy, wfrag[kc], h3);
            h4 = WMMA_F32X4(axy, wfrag[kc], h4);
        }
        // D layout: lane holds rows m = rowbase + (lane>>4)*8 + v, col jrow.
        // Transposed lh makes these 8 consecutive floats.
        const int mb = rowbase + (lane >> 4) * 8;
        store_frag_rows(&lh[0][jrow][mb], h0);
        store_frag_rows(&lh[1][jrow][mb], h1);
        store_frag_rows(&lh[2][jrow][mb], h2);
        store_frag_rows(&lh[3][jrow][mb], h3);
        store_frag_rows(&lh[4][jrow][mb], h4);
    }
    __syncthreads();   // single-wave WG: cheap LDS write->read fence

    // ---- vertical pass: banded matrix as A, H as B; 5 WMMAs per wait ----
    v8f o0 = {}, o1 = {}, o2 = {}, o3 = {}, o4 = {};
#pragma unroll
    for (int kc = 0; kc < CHUNKS; ++kc) {
        int k0 = 4 * kc + khalf;
        v2f b0 = *(const v2f*)&lh[0][jrow][k0];
        v2f b1 = *(const v2f*)&lh[1][jrow][k0];
        v2f b2 = *(const v2f*)&lh[2][jrow][k0];
        v2f b3 = *(const v2f*)&lh[3][jrow][k0];
        v2f b4 = *(const v2f*)&lh[4][jrow][k0];
        o0 = WMMA_F32X4(wfrag[kc], b0, o0);
        o1 = WMMA_F32X4(wfrag[kc], b1, o1);
        o2 = WMMA_F32X4(wfrag[kc], b2, o2);
        o3 = WMMA_F32X4(wfrag[kc], b3, o3);
        o4 = WMMA_F32X4(wfrag[kc], b4, o4);
    }

    // ---- pointwise SSIM map + masked tile sum ----
    const float C1 = 1.0e-4f;   // (0.01*1.0)^2
    const float C2 = 9.0e-4f;   // (0.03*1.0)^2
    float lsum = 0.0f;
#pragma unroll
    for (int v = 0; v < 8; ++v) {
        int m    = v + (lane >> 4) * 8;
        int orow = r0 + m;
        int ocol = c0 + jrow;
        if (orow < OUT_H && ocol < OUT_W) {
            float mu1 = o0[v], mu2 = o1[v];
            float exx = o2[v], eyy = o3[v], exy = o4[v];
            float s1  = exx - mu1 * mu1;
            float s2  = eyy - mu2 * mu2;
            float s12 = exy - mu1 * mu2;
            float cs  = (2.0f * s12 + C2) / (s1 + s2 + C2);
            float lum = (2.0f * mu1 * mu2 + C1) / (mu1 * mu1 + mu2 * mu2 + C1);
            lsum += lum * cs;
        }
    }
    // wave32 butterfly reduction
#pragma unroll
    for (int off = 16; off > 0; off >>= 1)
        lsum += __shfl_xor(lsum, off, 32);
    if (lane == 0) partial[bid] = lsum;
}

__global__ __launch_bounds__(256)
void ssim_reduce_kernel(const float* __restrict__ partial, float* __restrict__ out) {
    __shared__ double sm[256];
    double s = 0.0;
    for (int i = threadIdx.x; i < NTILES; i += 256) s += (double)partial[i];
    sm[threadIdx.x] = s;
    __syncthreads();
    for (int off = 128; off > 0; off >>= 1) {
        if (threadIdx.x < off) sm[threadIdx.x] += sm[threadIdx.x + off];
        __syncthreads();
    }
    if (threadIdx.x == 0) {
        const double npix = (double)NIMG * OUT_H * OUT_W;  // 8,064,128
        out[0] = (float)(1.0 - sm[0] / npix);
    }
}

extern "C" void kernel_launch(void* const* d_in, const int* in_sizes, int n_in,
                              void* d_out, int out_size, void* d_ws, size_t ws_size,
                              hipStream_t stream) {
    (void)in_sizes; (void)n_in; (void)out_size; (void)ws_size;
    const float* y_pred = (const float*)d_in[0];
    const float* y_true = (const float*)d_in[1];
    float* partial = (float*)d_ws;          // needs NTILES*4 = 128 KB scratch
    float* out     = (float*)d_out;

    ssim_tile_kernel<<<NTILES, 32, 0, stream>>>(y_pred, y_true, partial);
    ssim_reduce_kernel<<<1, 256, 0, stream>>>(partial, out);
}